// stgru_19550691131756
// MI455X (gfx1250) — compile-verified
//
#include <hip/hip_runtime.h>
#include <stdint.h>

// ---------------------------------------------------------------------------
// STGRU fused step kernel for MI455X (gfx1250).
// HBM-bound (~0.4 GB traffic, ~1 GFLOP over 3 steps => ~17us roofline floor
// at 23.3 TB/s). One fused pass per time step; h staged through d_ws because
// the next step's warp gathers h at arbitrary pixels (global sync needed).
//
// CDNA5 usage: Tensor Data Mover (tensor_load_to_lds, ISA ch.8) DMAs the
// structured halo tiles (I_last 34x10x3, unary 34x10) into packed LDS while
// the vector pipes run the data-dependent bilinear gathers; wave 0 then
// s_wait_tensorcnt(0) before the single workgroup barrier. WMMA deliberately
// unused: 3x3 convs with 1-3 input channels (K=9..27) are <6% MACC
// utilization and the VALU FMAs are free under the memory roof.
// ---------------------------------------------------------------------------

#define TX 32
#define TY 8
#define TW (TX + 2)          // 34  halo tile width
#define THL (TY + 2)         // 10  halo tile height
#define NS (TW * THL)        // 340 halo slots
#define NTHREADS 256

typedef uint32_t u32;
typedef uint64_t u64;
typedef unsigned int v4u __attribute__((ext_vector_type(4)));
typedef int          v4i __attribute__((ext_vector_type(4)));
typedef int          v8i __attribute__((ext_vector_type(8)));

// Build a TDM descriptor (D#) per CDNA5 ISA ch.8 and issue TENSOR_LOAD_TO_LDS.
// Tile: tile0 x tile1 x tile2 elements (4B each), row stride stride0,
// plane stride stride1 (elements). Packed contiguously into LDS at lds_byte_off.
__device__ __forceinline__ void tdm_load_tile_to_lds(
    const float* gptr, u32 lds_byte_off,
    u32 tdim0, u32 tdim1, u32 tdim2,
    u32 tile0, u32 tile1, u32 tile2,
    u32 stride0, u32 stride1)
{
  u64 ga = (u64)(uintptr_t)gptr;
  v4u g0;
  g0.x = 1u;                                               // count=1, gather off
  g0.y = lds_byte_off;                                     // lds_addr [63:32]
  g0.z = (u32)ga;                                          // global_addr[31:0]
  g0.w = ((u32)(ga >> 32) & 0x01FFFFFFu) | (2u << 30);     // addr[56:32] | type=2
  v8i g1;
  g1[0] = (int)(2u << 16);                                 // data_size=2 (4 bytes)
  g1[1] = (int)((tdim0 & 0xFFFFu) << 16);                  // tensor_dim0[15:0]
  g1[2] = (int)(((tdim0 >> 16) & 0xFFFFu) |                // tensor_dim0[31:16]
                ((tdim1 & 0xFFFFu) << 16));                // tensor_dim1[15:0]
  g1[3] = (int)(((tdim1 >> 16) & 0xFFFFu) |                // tensor_dim1[31:16]
                ((tile0 & 0xFFFFu) << 16));                // tile_dim0
  g1[4] = (int)((tile1 & 0xFFFFu) | ((tile2 & 0xFFFFu) << 16)); // tile_dim1/2
  g1[5] = (int)stride0;                                    // tensor_dim0_stride lo
  g1[6] = (int)((stride1 & 0xFFFFu) << 16);                // stride0 hi=0 | stride1 lo16
  g1[7] = (int)(stride1 >> 16);                            // stride1[47:16]
  v4i g2; g2.x = (int)tdim2; g2.y = 0; g2.z = 0; g2.w = 0; // tensor_dim2
  v4i g3; g3.x = 0; g3.y = 0; g3.z = 0; g3.w = 0;
#if defined(__clang_major__) && (__clang_major__ >= 23)
  v8i z8 = {0, 0, 0, 0, 0, 0, 0, 0};
  __builtin_amdgcn_tensor_load_to_lds(g0, g1, g2, g3, z8, 0);
#else
  __builtin_amdgcn_tensor_load_to_lds(g0, g1, g2, g3, 0);
#endif
}

__device__ __forceinline__ int iclamp(int v, int lo, int hi) {
  return v < lo ? lo : (v > hi ? hi : v);
}

__global__ __launch_bounds__(NTHREADS) void stgru_step_kernel(
    const float* __restrict__ img_i,     // [B,3,H,W] raw 0..255
    const float* __restrict__ img_last,  // [B,3,H,W] raw 0..255
    const float* __restrict__ flow,      // [B,2,H,W] ch0=dy ch1=dx
    const float* __restrict__ x_next,    // [B,1,H,W]
    const float* __restrict__ h_prev,    // [B,1,H,W]
    const float* __restrict__ w_ir,      // 27 floats [c][ky][kx]
    const float* __restrict__ w_xh,      // 9 floats
    const float* __restrict__ w_hh,      // 9 floats
    float* __restrict__ h_out,           // [B,1,H,W]
    int H, int W)
{
  __shared__ float s_last[3 * NS];   // I_last tile (raw scale), TDM-filled
  __shared__ float s_x[NS];          // unary input tile, TDM-filled
  __shared__ float s_warp[3 * NS];   // warped I (raw scale)
  __shared__ float s_uw[NS];         // warped h

  const int tid = (int)threadIdx.x;
  const int x0 = (int)blockIdx.x * TX;
  const int y0 = (int)blockIdx.y * TY;
  const int b  = (int)blockIdx.z;
  const int HW = H * W;

  const float* ilast_b = img_last + (size_t)b * 3 * HW;
  const float* flow_b  = flow     + (size_t)b * 2 * HW;
  const float* xn_b    = x_next   + (size_t)b * HW;
  const float* img_b   = img_i    + (size_t)b * 3 * HW;
  const float* hp_b    = h_prev   + (size_t)b * HW;

  // Halo fully inside the image? (block-uniform, scalar branch)
  const bool interior = (x0 >= 1) && (y0 >= 1) &&
                        (x0 + TX + 1 <= W) && (y0 + TY + 1 <= H);
  const int wid = __builtin_amdgcn_readfirstlane((int)(threadIdx.x >> 5));

  if (interior) {
    // --- CDNA5 TDM: issue async tensor DMAs; completion awaited AFTER the
    //     gather phase so the DMA rides under ~5k data-dependent loads. -----
    if (wid == 0) {
      const size_t to = (size_t)(y0 - 1) * W + (x0 - 1);
      tdm_load_tile_to_lds(ilast_b + to, (u32)(uintptr_t)&s_last[0],
                           (u32)W, (u32)H, 3u, TW, THL, 3u, (u32)W, (u32)HW);
      tdm_load_tile_to_lds(xn_b + to,    (u32)(uintptr_t)&s_x[0],
                           (u32)W, (u32)H, 1u, TW, THL, 0u, (u32)W, (u32)HW);
    }
  } else {
    // --- edge blocks: guarded loads with SAME zero padding -----------------
    for (int s = tid; s < NS; s += NTHREADS) {
      const int ry = s / TW, rx = s - ry * TW;
      const int py = y0 - 1 + ry, px = x0 - 1 + rx;
      const bool in = ((unsigned)px < (unsigned)W) && ((unsigned)py < (unsigned)H);
      const size_t o = (size_t)(in ? py : 0) * W + (in ? px : 0);
      s_last[0 * NS + s] = in ? ilast_b[o]          : 0.f;
      s_last[1 * NS + s] = in ? ilast_b[HW + o]     : 0.f;
      s_last[2 * NS + s] = in ? ilast_b[2 * HW + o] : 0.f;
      s_x[s]             = in ? xn_b[o]             : 0.f;
    }
  }

  // --- Phase 2: bilinear warp gathers into halo tiles (border clamp). ------
  // Flow is read register-direct (each slot consumed by exactly one thread).
  const float scy = (float)H / (float)(H - 1);
  const float scx = (float)W / (float)(W - 1);
  for (int s = tid; s < NS; s += NTHREADS) {
    const int ry = s / TW, rx = s - ry * TW;
    const int py = y0 - 1 + ry, px = x0 - 1 + rx;
    float v0 = 0.f, v1 = 0.f, v2 = 0.f, uw = 0.f;
    if (((unsigned)px < (unsigned)W) && ((unsigned)py < (unsigned)H)) {
      const size_t fo = (size_t)py * W + px;
      const float dy = flow_b[fo];
      const float dx = flow_b[HW + fo];
      const float yy = ((float)py + dy) * scy - 0.5f;
      const float xx = ((float)px + dx) * scx - 0.5f;
      const float yf = floorf(yy), xf = floorf(xx);
      const float wy = yy - yf,    wx = xx - xf;
      const int iy = (int)yf, ix = (int)xf;
      const int cy0 = iclamp(iy,     0, H - 1), cy1 = iclamp(iy + 1, 0, H - 1);
      const int cx0 = iclamp(ix,     0, W - 1), cx1 = iclamp(ix + 1, 0, W - 1);
      const float w00 = (1.f - wy) * (1.f - wx), w01 = (1.f - wy) * wx;
      const float w10 = wy * (1.f - wx),         w11 = wy * wx;
      const int o00 = cy0 * W + cx0, o01 = cy0 * W + cx1;
      const int o10 = cy1 * W + cx0, o11 = cy1 * W + cx1;
      const float* p0 = img_b;
      const float* p1 = img_b + HW;
      const float* p2 = img_b + 2 * HW;
      v0 = p0[o00] * w00 + p0[o01] * w01 + p0[o10] * w10 + p0[o11] * w11;
      v1 = p1[o00] * w00 + p1[o01] * w01 + p1[o10] * w10 + p1[o11] * w11;
      v2 = p2[o00] * w00 + p2[o01] * w01 + p2[o10] * w10 + p2[o11] * w11;
      uw = hp_b[o00] * w00 + hp_b[o01] * w01 + hp_b[o10] * w10 + hp_b[o11] * w11;
    }
    s_warp[0 * NS + s] = v0;
    s_warp[1 * NS + s] = v1;
    s_warp[2 * NS + s] = v2;
    s_uw[s] = uw;
  }

  // Issuing wave drains the TDM before signaling the (single) barrier.
  if (interior && wid == 0)
    __builtin_amdgcn_s_wait_tensorcnt(0);
  __syncthreads();

  // --- Phase 3: 3x3 convs from LDS (diff computed on the fly) + gate -------
  const int tx = tid & (TX - 1);
  const int ty = tid >> 5;
  float accR = 0.f, accX = 0.f, accH = 0.f;
#pragma unroll
  for (int ky = 0; ky < 3; ++ky) {
#pragma unroll
    for (int kx = 0; kx < 3; ++kx) {
      const int ss = (ty + ky) * TW + (tx + kx);
      const int k = ky * 3 + kx;
      // |I_last - I_warp| at raw 0..255 scale; 1/255 hoisted onto accR below.
      accR = fmaf(fabsf(s_last[0 * NS + ss] - s_warp[0 * NS + ss]), w_ir[k],      accR);
      accR = fmaf(fabsf(s_last[1 * NS + ss] - s_warp[1 * NS + ss]), w_ir[9 + k],  accR);
      accR = fmaf(fabsf(s_last[2 * NS + ss] - s_warp[2 * NS + ss]), w_ir[18 + k], accR);
      accX = fmaf(s_x[ss],  w_xh[k], accX);
      accH = fmaf(s_uw[ss], w_hh[k], accH);
    }
  }
  // r = 1 - sigmoid(v) = 1 / (1 + e^v)
  const float r = 1.f / (1.f + __expf(accR * (1.f / 255.f)));
  const int opy = y0 + ty, opx = x0 + tx;
  if (opy < H && opx < W)
    h_out[(size_t)b * HW + (size_t)opy * W + opx] = accX + accH * r;
}

extern "C" void kernel_launch(void* const* d_in, const int* in_sizes, int n_in,
                              void* d_out, int out_size, void* d_ws, size_t ws_size,
                              hipStream_t stream) {
  (void)in_sizes; (void)n_in; (void)out_size; (void)ws_size;
  const float* images = (const float*)d_in[0];  // [T,B,3,H,W]
  const float* flows  = (const float*)d_in[1];  // [T-1,B,2,H,W]
  const float* unary  = (const float*)d_in[2];  // [T,B,1,H,W]
  const float* W_ir   = (const float*)d_in[3];  // [T-1,1,3,3,3]
  const float* W_xh   = (const float*)d_in[4];  // [T-1,1,1,3,3]
  const float* W_hh   = (const float*)d_in[5];  // [T-1,1,1,3,3]
  float* out = (float*)d_out;

  const int T = 4, B = 8, H = 480, W = 864;
  const size_t HW    = (size_t)H * W;
  const size_t img_t = (size_t)B * 3 * HW;
  const size_t flo_t = (size_t)B * 2 * HW;
  const size_t un_t  = (size_t)B * HW;

  float* h1 = (float*)d_ws;            // step 0 output
  float* h2 = h1 + un_t;               // step 1 output (needs 2*B*H*W*4 B ws)
  const float* img_last = images + (size_t)(T - 1) * img_t;

  dim3 grid((W + TX - 1) / TX, (H + TY - 1) / TY, B);
  dim3 block(NTHREADS);

  const float* hprev = unary;          // h = unary_inputs[0]
  float* houts[3] = {h1, h2, out};
  for (int i = 0; i < T - 1; ++i) {
    stgru_step_kernel<<<grid, block, 0, stream>>>(
        images + (size_t)i * img_t, img_last,
        flows + (size_t)i * flo_t,
        unary + (size_t)(i + 1) * un_t,
        hprev,
        W_ir + (size_t)i * 27, W_xh + (size_t)i * 9, W_hh + (size_t)i * 9,
        houts[i], H, W);
    hprev = houts[i];
  }
}